// PatchNet_ms_66855460929918
// MI455X (gfx1250) — compile-verified
//
#include <hip/hip_runtime.h>
#include <hip/hip_bf16.h>
#include <math.h>

// ---------------------------------------------------------------------------
// PatchNet: pooling -> LN -> MLP(gelu) -> score -> perturbed top-k -> gather
// Shapes fixed by the reference: B=8, C=64, T=64, H=W=56, E=128, K=8, NS=500
// ---------------------------------------------------------------------------
#define B_    8
#define C_    64
#define T_    64
#define H_    56
#define W_    56
#define HW_   (H_ * W_)      /* 3136 */
#define E_    128
#define K_    8
#define NS_   500
#define SIGMA_ 0.05f
#define LN_EPS_ 1e-5f

typedef __attribute__((ext_vector_type(2))) float v2f;
typedef __attribute__((ext_vector_type(8))) float v8f;

__device__ __forceinline__ float gelu_exact(float v) {
    return 0.5f * v * (1.0f + erff(v * 0.70710678118654752f));
}

// ---------------------------------------------------------------------------
// Kernel 1: avg/max pool over H*W for each (b,c,t).  Reads all of x (411 MB).
// x layout: (B,C,T,H,W) contiguous -> block bid owns x[bid*HW .. +HW)
// feat layout: (B*T, E) rows; feat[row][c]=avg, feat[row][C+c]=max
// ---------------------------------------------------------------------------
__global__ void k_pool(const float* __restrict__ x, float* __restrict__ feat) {
    __shared__ float ssum[256];
    __shared__ float smax[256];
    const int bid = blockIdx.x;                 // (b*C + c)*T + t
    const float* xs = x + (size_t)bid * HW_;
    float s = 0.0f, m = -INFINITY;
    for (int i = threadIdx.x; i < HW_; i += 256) {
        float v = xs[i];
        s += v;
        m = fmaxf(m, v);
    }
    ssum[threadIdx.x] = s;
    smax[threadIdx.x] = m;
    __syncthreads();
    for (int off = 128; off > 0; off >>= 1) {
        if ((int)threadIdx.x < off) {
            ssum[threadIdx.x] += ssum[threadIdx.x + off];
            smax[threadIdx.x] = fmaxf(smax[threadIdx.x], smax[threadIdx.x + off]);
        }
        __syncthreads();
    }
    if (threadIdx.x == 0) {
        int t = bid % T_;
        int c = (bid / T_) % C_;
        int b = bid / (T_ * C_);
        float* row = feat + (size_t)(b * T_ + t) * E_;
        row[c]      = ssum[0] * (1.0f / (float)HW_);
        row[C_ + c] = smax[0];
    }
}

// ---------------------------------------------------------------------------
// Kernel 2a: LayerNorm over E per row (512 rows).
// ---------------------------------------------------------------------------
__global__ void k_lnorm(const float* __restrict__ feat, const float* __restrict__ ln_w,
                        const float* __restrict__ ln_b, float* __restrict__ normf) {
    __shared__ float red[128];
    const int row = blockIdx.x;
    const int e = threadIdx.x;             // 128 threads
    float v = feat[(size_t)row * E_ + e];
    red[e] = v;
    __syncthreads();
    for (int off = 64; off > 0; off >>= 1) {
        if (e < off) red[e] += red[e + off];
        __syncthreads();
    }
    float mu = red[0] * (1.0f / (float)E_);
    __syncthreads();
    float d = v - mu;
    red[e] = d * d;
    __syncthreads();
    for (int off = 64; off > 0; off >>= 1) {
        if (e < off) red[e] += red[e + off];
        __syncthreads();
    }
    float var = red[0] * (1.0f / (float)E_);
    float nv = d * rsqrtf(var + LN_EPS_);
    normf[(size_t)row * E_ + e] = nv * ln_w[e] + ln_b[e];
}

// ---------------------------------------------------------------------------
// Kernel 2b: h = gelu(normf @ w1 + b1), (512x128)@(128x128), via
// V_WMMA_F32_16X16X4_F32.  One wave per 16x16 tile; 32 k-steps of 4.
// A 16x4 layout: M=lane%16; VGPR0 holds K={0,2}, VGPR1 K={1,3} (per half-wave)
// B 4x16 layout: N=lane%16; VGPR0 holds K={0,2}, VGPR1 K={1,3}
// D: VGPR r -> (M=r, lanes 0-15) / (M=r+8, lanes 16-31)
// ---------------------------------------------------------------------------
__global__ void k_mlp1(const float* __restrict__ normf, const float* __restrict__ w1,
                       const float* __restrict__ b1, float* __restrict__ h) {
    const int wave = blockIdx.x * 4 + (threadIdx.x >> 5);   // 256 waves
    const int lane = threadIdx.x & 31;
    const int mt = wave >> 3;            // 0..31  (row tiles, 512/16)
    const int nt = wave & 7;             // 0..7   (col tiles, 128/16)
    const int row0 = mt * 16, col0 = nt * 16;
    const int nm = lane & 15;
    const int half = lane >> 4;

    v8f acc = {};
    for (int kk = 0; kk < 32; ++kk) {
        int ka = kk * 4 + half * 2;
        v2f a, bm;
        const float* ap = normf + (size_t)(row0 + nm) * E_ + ka;
        a.x = ap[0];
        a.y = ap[1];
        const float* bp = w1 + (size_t)ka * E_ + col0 + nm;
        bm.x = bp[0];
        bm.y = bp[E_];
        acc = __builtin_amdgcn_wmma_f32_16x16x4_f32(false, a, false, bm,
                                                    (short)0, acc, false, false);
    }
    const int col = col0 + nm;
    const float bb = b1[col];
    for (int r = 0; r < 8; ++r) {
        int row = row0 + r + half * 8;
        h[(size_t)row * E_ + col] = gelu_exact(acc[r] + bb);
    }
}

// ---------------------------------------------------------------------------
// Kernel 3: per-batch scores.  local = h[...,:64]; global = mean_t h[...,64:]
// s = gelu(local.w2[:64] + global.w2[64:] + b2), then min-max norm over T.
// One block of 64 threads per b (thread == t == global-channel j).
// ---------------------------------------------------------------------------
__global__ void k_scores(const float* __restrict__ h, const float* __restrict__ w2,
                         const float* __restrict__ b2, float* __restrict__ scores) {
    __shared__ float g[C_];
    __shared__ float sv[T_];
    __shared__ float mmn[2];
    const int b = blockIdx.x;
    const int tid = threadIdx.x;           // 64 threads
    // global mean over t of h[b, :, 64 + tid]
    float gs = 0.0f;
    for (int t = 0; t < T_; ++t)
        gs += h[(size_t)(b * T_ + t) * E_ + C_ + tid];
    g[tid] = gs * (1.0f / (float)T_);
    __syncthreads();
    // per-t score
    float dot = 0.0f;
    const float* hr = h + (size_t)(b * T_ + tid) * E_;
    for (int j = 0; j < C_; ++j) dot += hr[j] * w2[j];
    float gdot = 0.0f;
    for (int j = 0; j < C_; ++j) gdot += g[j] * w2[C_ + j];
    float s = gelu_exact(dot + gdot + b2[0]);
    sv[tid] = s;
    __syncthreads();
    if (tid == 0) {
        float lo = sv[0], hi = sv[0];
        for (int t = 1; t < T_; ++t) {
            lo = fminf(lo, sv[t]);
            hi = fmaxf(hi, sv[t]);
        }
        mmn[0] = lo;
        mmn[1] = hi;
    }
    __syncthreads();
    scores[b * T_ + tid] = (s - mmn[0]) / (mmn[1] - mmn[0] + 1e-5f);
}

// ---------------------------------------------------------------------------
// Kernel 4: perturbed top-k indicator.  Noise reproduces JAX threefry2x32
// (key data = {0,42}, counter = iota(B*NS*T) split in halves) and
// normal = sqrt(2)*erfinv(uniform(nextafter(-1,0), 1)).
// One block of 512 threads per b; thread = noise sample ns.
// ---------------------------------------------------------------------------
__device__ __forceinline__ void threefry2x32(uint32_t k0, uint32_t k1,
                                             uint32_t x0, uint32_t x1,
                                             uint32_t& o0, uint32_t& o1) {
    const uint32_t ks0 = k0, ks1 = k1, ks2 = 0x1BD11BDAu ^ k0 ^ k1;
    x0 += ks0; x1 += ks1;
#define TFR(r) { x0 += x1; x1 = (x1 << (r)) | (x1 >> (32 - (r))); x1 ^= x0; }
    TFR(13) TFR(15) TFR(26) TFR(6)
    x0 += ks1; x1 += ks2 + 1u;
    TFR(17) TFR(29) TFR(16) TFR(24)
    x0 += ks2; x1 += ks0 + 2u;
    TFR(13) TFR(15) TFR(26) TFR(6)
    x0 += ks0; x1 += ks1 + 3u;
    TFR(17) TFR(29) TFR(16) TFR(24)
    x0 += ks1; x1 += ks2 + 4u;
    TFR(13) TFR(15) TFR(26) TFR(6)
    x0 += ks2; x1 += ks0 + 5u;
#undef TFR
    o0 = x0; o1 = x1;
}

__device__ __forceinline__ float jax_normal_at(uint32_t i) {
    const uint32_t HALF = (uint32_t)(B_ * NS_ * T_) / 2u;   // 128000
    uint32_t x0, x1;
    const bool second = (i >= HALF);
    if (second) { x0 = i - HALF; x1 = i; }
    else        { x0 = i;        x1 = i + HALF; }
    uint32_t o0, o1;
    threefry2x32(0u, 42u, x0, x1, o0, o1);
    uint32_t bits = second ? o1 : o0;
    float f = __uint_as_float((bits >> 9) | 0x3F800000u) - 1.0f;   // [0,1)
    const float lo = -0.99999994f;                                  // nextafter(-1,0)
    float u = fmaxf(lo, fmaf(f, 1.0f - lo, lo));
    return 1.4142135623730951f * erfinvf(u);
}

__global__ void k_indicator(const float* __restrict__ scores, float* __restrict__ ind) {
    __shared__ float accs[K_ * T_];
    const int b = blockIdx.x;
    const int tid = threadIdx.x;           // 512 threads, 500 active samples
    if (tid < K_ * T_) accs[tid] = 0.0f;
    __syncthreads();
    if (tid < NS_) {
        float p[T_];
        const float* sr = scores + b * T_;
        const uint32_t base = (uint32_t)((b * NS_ + tid) * T_);
        for (int t = 0; t < T_; ++t)
            p[t] = sr[t] + SIGMA_ * jax_normal_at(base + t);
        int idxs[K_];
        for (int k = 0; k < K_; ++k) {         // top-k, ties -> lowest index
            int best = 0; float bv = p[0];
            for (int t = 1; t < T_; ++t)
                if (p[t] > bv) { bv = p[t]; best = t; }
            idxs[k] = best;
            p[best] = -INFINITY;
        }
        for (int i = 1; i < K_; ++i) {          // sort indices ascending
            int v = idxs[i], j = i - 1;
            while (j >= 0 && idxs[j] > v) { idxs[j + 1] = idxs[j]; --j; }
            idxs[j + 1] = v;
        }
        for (int k = 0; k < K_; ++k)
            atomicAdd(&accs[k * T_ + idxs[k]], 1.0f / (float)NS_);
    }
    __syncthreads();
    if (tid < K_ * T_) ind[b * (K_ * T_) + tid] = accs[tid];
}

// ---------------------------------------------------------------------------
// Kernel 5: the bandwidth-dominant gather, as a WMMA GEMM per (b,c):
//   out[b,c,k,hw] = sum_t ind[b,k,t] * x[b,c,t,hw]    (M=8 pad 16, K=64, N=3136)
// One wave per 16-column tile; x is read exactly once (411 MB), out 51 MB.
// ---------------------------------------------------------------------------
__global__ void k_gather(const float* __restrict__ x, const float* __restrict__ ind,
                         float* __restrict__ out) {
    const int NTILE = HW_ / 16;                        // 196
    const int wave = blockIdx.x * 4 + (threadIdx.x >> 5);
    const int lane = threadIdx.x & 31;
    const int b   = wave / (C_ * NTILE);
    const int rr0 = wave % (C_ * NTILE);
    const int ch  = rr0 / NTILE;
    const int col0 = (rr0 % NTILE) * 16;
    const int nm = lane & 15;
    const int half = lane >> 4;

    // Preload A = indicator[b] as 16 k-steps (rows 8..15 zero-padded)
    v2f A[16];
    const float* ib = ind + b * (K_ * T_);
    for (int kk = 0; kk < 16; ++kk) {
        int ka = kk * 4 + half * 2;
        if (nm < K_) { A[kk].x = ib[nm * T_ + ka]; A[kk].y = ib[nm * T_ + ka + 1]; }
        else         { A[kk].x = 0.0f;             A[kk].y = 0.0f; }
    }

    const float* xs = x + (size_t)(b * C_ + ch) * T_ * HW_;
    v8f acc = {};
    for (int kk = 0; kk < 16; ++kk) {
        int t0 = kk * 4 + half * 2;
        const float* bp = xs + (size_t)t0 * HW_ + col0 + nm;
        v2f bm;
        bm.x = bp[0];
        bm.y = bp[HW_];
        if (kk + 1 < 16)
            __builtin_prefetch(bp + 4 * HW_, 0, 1);   // global_prefetch_b8
        acc = __builtin_amdgcn_wmma_f32_16x16x4_f32(false, A[kk], false, bm,
                                                    (short)0, acc, false, false);
    }
    if (half == 0) {                    // only rows M=0..7 are valid output
        float* ob = out + (size_t)(b * C_ + ch) * K_ * HW_ + col0 + nm;
        for (int r = 0; r < K_; ++r)
            ob[(size_t)r * HW_] = acc[r];
    }
}

// ---------------------------------------------------------------------------
// Launch
// ---------------------------------------------------------------------------
extern "C" void kernel_launch(void* const* d_in, const int* in_sizes, int n_in,
                              void* d_out, int out_size, void* d_ws, size_t ws_size,
                              hipStream_t stream) {
    (void)in_sizes; (void)n_in; (void)out_size; (void)ws_size;
    const float* x   = (const float*)d_in[0];
    const float* lnw = (const float*)d_in[1];
    const float* lnb = (const float*)d_in[2];
    const float* w1  = (const float*)d_in[3];
    const float* b1  = (const float*)d_in[4];
    const float* w2  = (const float*)d_in[5];
    const float* b2  = (const float*)d_in[6];
    // d_in[7..9] = H, W, T scalars -> compile-time constants here
    float* out = (float*)d_out;

    char* ws = (char*)d_ws;
    float* feat   = (float*)(ws);              // 512*128 f32 = 256 KB
    float* normf  = (float*)(ws + 262144);     // 256 KB
    float* h      = (float*)(ws + 524288);     // 256 KB
    float* scores = (float*)(ws + 786432);     // 512 f32
    float* ind    = (float*)(ws + 790528);     // 8*8*64 f32

    k_pool     <<<B_ * C_ * T_, 256, 0, stream>>>(x, feat);
    k_lnorm    <<<B_ * T_,      128, 0, stream>>>(feat, lnw, lnb, normf);
    k_mlp1     <<<64,           128, 0, stream>>>(normf, w1, b1, h);
    k_scores   <<<B_,            64, 0, stream>>>(h, w2, b2, scores);
    k_indicator<<<B_,           512, 0, stream>>>(scores, ind);
    k_gather   <<<(B_ * C_ * (HW_ / 16)) / 4, 128, 0, stream>>>(x, ind, out);
}